// EnergyModelAdapter_59296318489074
// MI455X (gfx1250) — compile-verified
//
#include <hip/hip_runtime.h>
#include <math.h>

#define F_DIMC 128
#define H_DIMC 64
#define N_SPEC 4
#define TILE_M 32                   // atoms per wave (two 16-row WMMA tiles)
#define FSTRIDE 132                 // 128 + 4 pad: float4-aligned, bank-rotating
#define HSTRIDE 66                  // 64 + 2 pad: conflict-free lane*66+k reads

typedef __attribute__((ext_vector_type(2))) float v2f;
typedef __attribute__((ext_vector_type(8))) float v8f;

// ---------------- pass 1: counting sort by species ----------------
// ws layout (unsigned words): [0..3]=counts, [4..7]=offsets, [8..11]=cursors,
// [16..16+N) = sorted atom-index list.

__global__ void init_counters(unsigned* ws) {
    if (threadIdx.x < 12) ws[threadIdx.x] = 0u;
}

__global__ void count_species(const int* __restrict__ species,
                              unsigned* __restrict__ counts, int n) {
    int i = blockIdx.x * blockDim.x + threadIdx.x;
    if (i < n) atomicAdd(&counts[species[i] & 3], 1u);
}

__global__ void scan_counts(unsigned* ws) {
    if (threadIdx.x == 0 && blockIdx.x == 0) {
        unsigned off = 0;
        for (int s = 0; s < N_SPEC; ++s) {
            ws[4 + s] = off;   // segment start
            ws[8 + s] = off;   // scatter cursor
            off += ws[s];
        }
    }
}

__global__ void scatter_indices(const int* __restrict__ species,
                                unsigned* __restrict__ cursors,
                                unsigned* __restrict__ idxbuf, int n) {
    int i = blockIdx.x * blockDim.x + threadIdx.x;
    if (i < n) {
        unsigned p = atomicAdd(&cursors[species[i] & 3], 1u);
        idxbuf[p] = (unsigned)i;
    }
}

// ---------------- pass 2: WMMA MLP over 32-atom single-species tiles --------

__device__ __forceinline__ v8f wmma4(v2f a, v2f b, v8f c) {
    // D = A(16x4 f32) * B(4x16 f32) + C(16x16 f32)
    return __builtin_amdgcn_wmma_f32_16x16x4_f32(
        false, a, false, b, (short)0, c, false, false);
}

__device__ __forceinline__ float fast_tanh(float x) {
#if __has_builtin(__builtin_amdgcn_tanhf)
    return __builtin_amdgcn_tanhf(x);          // v_tanh_f32 (TRANS)
#else
    // branch-free: tanh(x) = 1 - 2/(exp(2x)+1); one v_exp_f32 + one v_rcp_f32
    float e = __builtin_amdgcn_exp2f(2.885390081777927f * x);  // exp(2x)
    return 1.0f - 2.0f * __builtin_amdgcn_rcpf(e + 1.0f);
#endif
}

__global__ __launch_bounds__(32) void mlp_tiles(
    const float* __restrict__ feats,
    const float* __restrict__ W1, const float* __restrict__ b1,
    const float* __restrict__ W2, const float* __restrict__ b2,
    const float* __restrict__ W3, const float* __restrict__ b3,
    const unsigned* __restrict__ counts, const unsigned* __restrict__ offs,
    const unsigned* __restrict__ idxbuf, float* __restrict__ E) {

    __shared__ float lA[TILE_M * FSTRIDE];     // feature tile (padded rows)
    __shared__ float lH[TILE_M * HSTRIDE];     // h1, then reused for h2
    __shared__ unsigned rows[TILE_M];

    const int s = blockIdx.y;
    const unsigned cnt = counts[s];
    const unsigned t0  = blockIdx.x * (unsigned)TILE_M;
    if (t0 >= cnt) return;                     // uniform exit, before any WMMA
    const unsigned base = offs[s] + t0;

    const int lane = threadIdx.x;              // wave32
    const int m    = lane & 15;                // row-in-tile / col-in-ntile
    const int hi   = lane >> 4;                // half-wave K split

    {
        unsigned slot = t0 + (unsigned)lane;
        rows[lane] = idxbuf[(slot < cnt) ? (base + (unsigned)lane) : base];
    }
    __syncthreads();

    // Gather 32 feature rows; one coalesced 512B row per iteration.
    for (int r = 0; r < TILE_M; ++r) {
        const float4* src = (const float4*)(feats + (size_t)rows[r] * F_DIMC);
        ((float4*)(lA + r * FSTRIDE))[lane] = src[lane];
    }
    __syncthreads();

    const float* W1s = W1 + (size_t)s * F_DIMC * H_DIMC;   // [128][64]
    const float* W2s = W2 + (size_t)s * H_DIMC * H_DIMC;   // [64][64]

    // ---- layer 1: [32x128] x [128x64]; B fragments reused across 2 M-tiles --
    v8f c0[4] = {v8f{}, v8f{}, v8f{}, v8f{}};
    v8f c1[4] = {v8f{}, v8f{}, v8f{}, v8f{}};
    for (int kk = 0; kk < F_DIMC / 4; ++kk) {
        const int kb = kk * 4 + hi * 2;
        v2f a0, a1;
        a0.x = lA[m * FSTRIDE + kb];        a0.y = lA[m * FSTRIDE + kb + 1];
        a1.x = lA[(16 + m) * FSTRIDE + kb]; a1.y = lA[(16 + m) * FSTRIDE + kb + 1];
        const float* w0 = W1s + (size_t)kb * H_DIMC;
        const float* w1 = W1s + (size_t)(kb + 1) * H_DIMC;
#pragma unroll
        for (int t = 0; t < 4; ++t) {
            v2f b;
            b.x = w0[t * 16 + m];
            b.y = w1[t * 16 + m];
            c0[t] = wmma4(a0, b, c0[t]);
            c1[t] = wmma4(a1, b, c1[t]);
        }
    }
    // bias + tanh; relayout D -> row-major h1 in LDS
#pragma unroll
    for (int t = 0; t < 4; ++t) {
        const int n = t * 16 + m;
        const float bias = b1[s * H_DIMC + n];
#pragma unroll
        for (int i = 0; i < 8; ++i) {
            const int mm = i + hi * 8;
            lH[mm * HSTRIDE + n]        = fast_tanh(c0[t][i] + bias);
            lH[(16 + mm) * HSTRIDE + n] = fast_tanh(c1[t][i] + bias);
        }
    }
    __syncthreads();

    // ---- layer 2: [32x64] x [64x64] ----
    v8f d0[4] = {v8f{}, v8f{}, v8f{}, v8f{}};
    v8f d1[4] = {v8f{}, v8f{}, v8f{}, v8f{}};
    for (int kk = 0; kk < H_DIMC / 4; ++kk) {
        const int kb = kk * 4 + hi * 2;
        v2f a0, a1;
        a0.x = lH[m * HSTRIDE + kb];        a0.y = lH[m * HSTRIDE + kb + 1];
        a1.x = lH[(16 + m) * HSTRIDE + kb]; a1.y = lH[(16 + m) * HSTRIDE + kb + 1];
        const float* w0 = W2s + (size_t)kb * H_DIMC;
        const float* w1 = W2s + (size_t)(kb + 1) * H_DIMC;
#pragma unroll
        for (int t = 0; t < 4; ++t) {
            v2f b;
            b.x = w0[t * 16 + m];
            b.y = w1[t * 16 + m];
            d0[t] = wmma4(a0, b, d0[t]);
            d1[t] = wmma4(a1, b, d1[t]);
        }
    }
    __syncthreads();
    // bias + tanh; h2 overwrites lH (all layer-2 reads already issued in-order)
#pragma unroll
    for (int t = 0; t < 4; ++t) {
        const int n = t * 16 + m;
        const float bias = b2[s * H_DIMC + n];
#pragma unroll
        for (int i = 0; i < 8; ++i) {
            const int mm = i + hi * 8;
            lH[mm * HSTRIDE + n]        = fast_tanh(d0[t][i] + bias);
            lH[(16 + mm) * HSTRIDE + n] = fast_tanh(d1[t][i] + bias);
        }
    }
    __syncthreads();

    // ---- layer 3: [32x64] x [64x1] as per-lane dot (N=1 wastes WMMA) ----
    {
        const float* w3 = W3 + (size_t)s * H_DIMC;
        float sum = b3[s];
#pragma unroll 8
        for (int k = 0; k < H_DIMC; ++k)
            sum = fmaf(lH[lane * HSTRIDE + k], w3[k], sum);
        if (t0 + (unsigned)lane < cnt) E[rows[lane]] = sum;
    }
}

// ---------------- launch ----------------

extern "C" void kernel_launch(void* const* d_in, const int* in_sizes, int n_in,
                              void* d_out, int out_size, void* d_ws, size_t ws_size,
                              hipStream_t stream) {
    const float* feats  = (const float*)d_in[0];
    const int*   specie = (const int*)d_in[1];
    const float* W1 = (const float*)d_in[2];
    const float* b1 = (const float*)d_in[3];
    const float* W2 = (const float*)d_in[4];
    const float* b2 = (const float*)d_in[5];
    const float* W3 = (const float*)d_in[6];
    const float* b3 = (const float*)d_in[7];
    float* E = (float*)d_out;

    const int n = in_sizes[1];                 // N atoms (species vector length)

    unsigned* ws      = (unsigned*)d_ws;
    unsigned* counts  = ws;
    unsigned* offsets = ws + 4;
    unsigned* cursors = ws + 8;
    unsigned* idxbuf  = ws + 16;               // n unsigned entries

    init_counters<<<1, 32, 0, stream>>>(ws);
    count_species<<<(n + 255) / 256, 256, 0, stream>>>(specie, counts, n);
    scan_counts<<<1, 32, 0, stream>>>(ws);
    scatter_indices<<<(n + 255) / 256, 256, 0, stream>>>(specie, cursors, idxbuf, n);

    const int tiles = (n + TILE_M - 1) / TILE_M;   // worst case: one species owns all
    dim3 grid(tiles, N_SPEC, 1);
    mlp_tiles<<<grid, 32, 0, stream>>>(feats, W1, b1, W2, b2, W3, b3,
                                       counts, offsets, idxbuf, E);
}